// GNNQNetwork_16088947490816
// MI455X (gfx1250) — compile-verified
//
#include <hip/hip_runtime.h>

typedef __attribute__((ext_vector_type(2))) float v2f;
typedef __attribute__((ext_vector_type(8))) float v8f;

#define FEAT 64

// ---------------- degree / normalization ----------------

__global__ void k_deg_init(float* deg, int n) {
    int i = blockIdx.x * blockDim.x + threadIdx.x;
    if (i < n) deg[i] = 1.0f;  // self-loop contribution
}

__global__ void k_deg_count(const int* __restrict__ dst, float* deg, int e) {
    int i = blockIdx.x * blockDim.x + threadIdx.x;
    if (i < e) unsafeAtomicAdd(&deg[dst[i]], 1.0f);  // hardware global_atomic_add_f32
}

__global__ void k_dinv(const float* __restrict__ deg, float* dinv, int n) {
    int i = blockIdx.x * blockDim.x + threadIdx.x;
    if (i < n) {
        float d = deg[i];
        dinv[i] = (d > 0.0f) ? rsqrtf(d) : 0.0f;
    }
}

// ---------------- fp32 WMMA GEMM: C[M,N] = A[M,K] @ B[K,N] ----------------
// Row-major. M multiple of 16, N multiple of 16, K multiple of 4.
// One wave computes one 16x16 output tile via V_WMMA_F32_16X16X4_F32.
// A 16x4 layout (ISA 7.12.2): lane l<16 row=l, VGPR0=K0 VGPR1=K1;
//                             lane l>=16 row=l-16, VGPR0=K2 VGPR1=K3.
// B 4x16 layout: VGPR0: lanes0-15 K0 / lanes16-31 K2; VGPR1: K1 / K3 (cols in lanes).
// C 16x16 f32: VGPR r: lanes0-15 row=r, lanes16-31 row=8+r.
__global__ void k_gemm_wmma_f32(const float* __restrict__ A,
                                const float* __restrict__ B,
                                float* __restrict__ C,
                                int M, int N, int K) {
    int wavesPerBlock = blockDim.x >> 5;
    int wave = blockIdx.x * wavesPerBlock + (threadIdx.x >> 5);
    int lane = threadIdx.x & 31;
    int tilesN = N >> 4;
    int tileM = (wave / tilesN) << 4;   // wave-uniform
    int tileN = (wave % tilesN) << 4;
    if (tileM >= M) return;             // wave-uniform exit: EXEC stays all-1s

    int half = lane >> 4;   // 0 | 1
    int l    = lane & 15;

    const float* arow = A + (size_t)(tileM + l) * K + (half << 1);
    const float* bcol = B + (size_t)(half << 1) * N + tileN + l;

    v8f acc = {};
    for (int k = 0; k < K; k += 4) {
        v2f a, b;
        a.x = arow[k + 0];
        a.y = arow[k + 1];
        b.x = bcol[(size_t)k * N];
        b.y = bcol[(size_t)(k + 1) * N];
        acc = __builtin_amdgcn_wmma_f32_16x16x4_f32(
            false, a, false, b, (short)0, acc, false, false);
    }

    float* crow = C + (size_t)(tileM + (half << 3)) * N + tileN + l;
#pragma unroll
    for (int r = 0; r < 8; ++r) crow[(size_t)r * N] = acc[r];
}

// ---------------- aggregation ----------------

// agg[i,:] = hlin[i,:] * dinv[i]^2   (self-loop term; also initializes buffer)
__global__ void k_agg_init(const float* __restrict__ hlin,
                           const float* __restrict__ dinv,
                           float* __restrict__ agg, int n) {
    int t = blockIdx.x * blockDim.x + threadIdx.x;  // n * (FEAT/4) threads
    int total = n * (FEAT / 4);
    if (t >= total) return;
    int i = t >> 4;                 // FEAT/4 == 16
    float s = dinv[i];
    s = s * s;
    float4 v = ((const float4*)hlin)[t];
    v.x *= s; v.y *= s; v.z *= s; v.w *= s;
    ((float4*)agg)[t] = v;
}

// per edge: agg[dst,:] += hlin[src,:] * dinv[src]*dinv[dst]
// 16 threads per edge, float4 gather + 4 hardware f32 atomics each.
__global__ void k_edge_scatter(const int* __restrict__ src,
                               const int* __restrict__ dst,
                               const float* __restrict__ dinv,
                               const float* __restrict__ hlin,
                               float* __restrict__ agg, int e) {
    int t = blockIdx.x * blockDim.x + threadIdx.x;
    int total = e * 16;
    if (t >= total) return;
    int eid = t >> 4;
    int f4  = t & 15;
    int s = src[eid];
    int d = dst[eid];
    float nrm = dinv[s] * dinv[d];
    float4 v = ((const float4*)(hlin + (size_t)s * FEAT))[f4];
    float* o = agg + (size_t)d * FEAT + (f4 << 2);
    unsafeAtomicAdd(o + 0, v.x * nrm);
    unsafeAtomicAdd(o + 1, v.y * nrm);
    unsafeAtomicAdd(o + 2, v.z * nrm);
    unsafeAtomicAdd(o + 3, v.w * nrm);
}

// h[i,f] = relu(agg[i,f] + b[f])
__global__ void k_bias_relu(const float* __restrict__ agg,
                            const float* __restrict__ b,
                            float* __restrict__ h, int n) {
    int t = blockIdx.x * blockDim.x + threadIdx.x;
    int total = n * FEAT;
    if (t >= total) return;
    float v = agg[t] + b[t & (FEAT - 1)];
    h[t] = v > 0.0f ? v : 0.0f;
}

// out[i,:2] = h[i,:64] @ Wl[64,2] + bl
__global__ void k_final_linear(const float* __restrict__ h,
                               const float* __restrict__ Wl,
                               const float* __restrict__ bl,
                               float* __restrict__ out, int n) {
    int i = blockIdx.x * blockDim.x + threadIdx.x;
    if (i >= n) return;
    float a0 = bl[0], a1 = bl[1];
    const float* hr = h + (size_t)i * FEAT;
#pragma unroll
    for (int k = 0; k < FEAT; ++k) {
        float v = hr[k];
        a0 += v * Wl[2 * k + 0];
        a1 += v * Wl[2 * k + 1];
    }
    out[2 * i + 0] = a0;
    out[2 * i + 1] = a1;
}

// ---------------- launch ----------------

extern "C" void kernel_launch(void* const* d_in, const int* in_sizes, int n_in,
                              void* d_out, int out_size, void* d_ws, size_t ws_size,
                              hipStream_t stream) {
    const float* x  = (const float*)d_in[0];
    const int*   ei = (const int*)d_in[1];
    const float* W1 = (const float*)d_in[2];
    const float* b1 = (const float*)d_in[3];
    const float* W2 = (const float*)d_in[4];
    const float* b2 = (const float*)d_in[5];
    const float* Wl = (const float*)d_in[6];
    const float* bl = (const float*)d_in[7];
    float* out = (float*)d_out;

    const int N = in_sizes[0] / 32;   // 50000 (multiple of 16)
    const int E = in_sizes[1] / 2;    // 800000
    const int* src = ei;
    const int* dst = ei + E;

    float* ws   = (float*)d_ws;
    int Npad    = (N + 3) & ~3;
    float* deg  = ws;
    float* dinv = ws + Npad;
    float* bufA = ws + 2 * (size_t)Npad;            // N*64 floats
    float* bufB = bufA + (size_t)N * FEAT;          // N*64 floats

    const int BT = 256;
    const int tiles   = (N / 16) * (FEAT / 16);
    const int gemmGrd = (tiles + 7) / 8;            // 8 waves (tiles) per 256-thr block
    const int nvGrd   = (N + BT - 1) / BT;
    const int neGrd   = (E + BT - 1) / BT;
    const int aggGrd  = (N * 16 + BT - 1) / BT;
    const int edgGrd  = (E * 16 + BT - 1) / BT;
    const int nfGrd   = (N * FEAT + BT - 1) / BT;

    // degrees + symmetric norm (shared by both layers)
    k_deg_init<<<nvGrd, BT, 0, stream>>>(deg, N);
    k_deg_count<<<neGrd, BT, 0, stream>>>(dst, deg, E);
    k_dinv<<<nvGrd, BT, 0, stream>>>(deg, dinv, N);

    // ---- layer 1: hlin1 = x @ W1 -> bufA ; agg -> bufB ; h1 -> bufA
    k_gemm_wmma_f32<<<gemmGrd, 256, 0, stream>>>(x, W1, bufA, N, FEAT, 32);
    k_agg_init<<<aggGrd, BT, 0, stream>>>(bufA, dinv, bufB, N);
    k_edge_scatter<<<edgGrd, BT, 0, stream>>>(src, dst, dinv, bufA, bufB, E);
    k_bias_relu<<<nfGrd, BT, 0, stream>>>(bufB, b1, bufA, N);

    // ---- layer 2: hlin2 = h1 @ W2 -> bufB ; agg -> bufA ; h2 -> bufB
    k_gemm_wmma_f32<<<gemmGrd, 256, 0, stream>>>(bufA, W2, bufB, N, FEAT, FEAT);
    k_agg_init<<<aggGrd, BT, 0, stream>>>(bufB, dinv, bufA, N);
    k_edge_scatter<<<edgGrd, BT, 0, stream>>>(src, dst, dinv, bufB, bufA, E);
    k_bias_relu<<<nfGrd, BT, 0, stream>>>(bufA, b2, bufB, N);

    // ---- head: out = h2 @ Wl + bl
    k_final_linear<<<nvGrd, BT, 0, stream>>>(bufB, Wl, bl, out, N);
}